// SpinSphericalFourierTransformer_25340307046999
// MI455X (gfx1250) — compile-verified
//
#include <hip/hip_runtime.h>
#include <hip/hip_bf16.h>
#include <math.h>

// ---------------------------------------------------------------------------
// Spin-weighted spherical harmonic transform (swsft_forward_with_symmetry)
// B=32, C=16, N=128 -> 512 images, L=63, SPIN=1.
//
// All-linear pipeline as fp32 WMMA GEMMs (v_wmma_f32_16x16x4_f32):
//   F^T  = (sph @ Bsel)^T                 (512*128 x 128) x (128 x 128pad)
//   Inm^T= (A_{parity(mm)} @ F[:,mm])^T   per image, (128pad x 128)x(128 x 64)
//   J[n] = Inm[n] + (-1)^mm Inm[127-n]    elementwise fold -> J[mm][img][n]
//   S    = J_mm @ D_mm^T                  per mm, (512x64)x(64x64), re & im
//   out  = fc * S
//
// All operand planes are padded to multiples of 16 and B-operands stored
// transposed so fragment loads are unconditional aligned 8-byte loads and
// accumulator stores are 16-byte stores (no divergent guards anywhere).
// ---------------------------------------------------------------------------

typedef float v2f __attribute__((ext_vector_type(2)));
typedef float v8f __attribute__((ext_vector_type(8)));

#define WMMA_F32(a, b, c) \
  __builtin_amdgcn_wmma_f32_16x16x4_f32(false, (a), false, (b), (short)0, (c), false, false)

#define PI_D 3.14159265358979323846

static constexpr int NIMG = 512;     // B*C
static constexpr int NN   = 128;     // sphere side
static constexpr int NM   = 127;     // valid mm columns (m = mm-63)
static constexpr int NMP  = 128;     // padded mm / n_idx dimension
static constexpr int NL   = 64;      // L+1
// workspace plane sizes (floats)
static constexpr size_t SZ_B  = (size_t)NMP * NN;          // 16384  Bsel^T[mm][x]
static constexpr size_t SZ_A  = (size_t)NMP * NN;          // 16384  A[n_idx][r]
static constexpr size_t SZ_DT = (size_t)NM * NL * NL;      // 520192 D^T[mm][l][n]
static constexpr size_t SZ_F  = (size_t)NIMG * NMP * NN;   // 8388608 F^T[img][mm][r]
static constexpr size_t SZ_I  = (size_t)NIMG * NMP * NMP;  // 8388608 Inm^T[img][mm][n]
static constexpr size_t SZ_J  = (size_t)NM * NIMG * NL;    // 4161536 J[mm][img][n]

// ---------------------------------------------------------------- precompute
__global__ void k_build_b(float* __restrict__ btr, float* __restrict__ bti) {
  int idx = blockIdx.x * blockDim.x + threadIdx.x;
  if (idx >= (int)SZ_B) return;
  int mm = idx >> 7, x = idx & 127;
  if (mm == 127) { btr[idx] = 0.f; bti[idx] = 0.f; return; }   // pad column
  int k = mm - 63;
  double ang = -2.0 * PI_D * (double)(k * x) / 128.0;
  btr[idx] = (float)cos(ang);
  bti[idx] = (float)sin(ang);
}

__global__ void k_build_a(const float* __restrict__ qw,
                          float* __restrict__ aer, float* __restrict__ aei,
                          float* __restrict__ aor_, float* __restrict__ aoi) {
  int idx = blockIdx.x * blockDim.x + threadIdx.x;
  if (idx >= (int)SZ_A) return;
  int ni = idx >> 7, r = idx & 127;
  if (ni == 127) { aer[idx] = aei[idx] = aor_[idx] = aoi[idx] = 0.f; return; } // pad row
  int f = (ni <= 63) ? ni : ni - 127;           // frequency of Inm row ni
  double scale = 2.0 * PI_D / 128.0;            // fft2 scale from reference
  double a0 = -2.0 * PI_D * (double)f * (double)r / 254.0;
  double a0r = (double)qw[r] * cos(a0), a0i = (double)qw[r] * sin(a0);
  double a1r = 0.0, a1i = 0.0;
  if (r >= 1 && r <= 126) {                     // mirrored torus rows 254-r
    double a1 = -2.0 * PI_D * (double)f * (double)(254 - r) / 254.0;
    a1r = (double)qw[254 - r] * cos(a1);
    a1i = (double)qw[254 - r] * sin(a1);
  }
  aer[idx]  = (float)(scale * (a0r + a1r));
  aei[idx]  = (float)(scale * (a0i + a1i));
  aor_[idx] = (float)(scale * (a0r - a1r));
  aoi[idx]  = (float)(scale * (a0i - a1i));
}

__global__ void k_build_dt(const float* __restrict__ wd, float* __restrict__ dt) {
  // dt[mm][l][n] = wd[l][63+n][mm] * wd[l][63+n][62]   (spin=1 -> col L-spin=62)
  int idx = blockIdx.x * blockDim.x + threadIdx.x;
  if (idx >= (int)SZ_DT) return;
  int mm = idx / (NL * NL);
  int rem = idx % (NL * NL);
  int l = rem >> 6, n = rem & 63;
  size_t base = (size_t)l * NM * NM + (size_t)(63 + n) * NM;
  dt[idx] = wd[base + mm] * wd[base + 62];
}

// ---------------------------------------------- stage 1: F^T = (sph @ Bsel)^T
__global__ void k_stage1(const float* __restrict__ sph,
                         const float* __restrict__ btr, const float* __restrict__ bti,
                         float* __restrict__ ftr, float* __restrict__ fti) {
  // grid.x = (512*128)/16 = 4096 M-tiles; 8 waves, wave w owns N-tile w (cols 16w..).
  int wave = threadIdx.x >> 5, lane = threadIdx.x & 31;
  int half = lane >> 4, lm = lane & 15;
  int m0 = blockIdx.x * 16;                    // global row = img*128 + r
  int col = wave * 16 + lm;                    // mm (0..127, padded)
  v8f accr = {}, acci = {};
  const float* arow = sph + (size_t)(m0 + lm) * NN;
  const float* brow = btr + (size_t)col * NN;  // Bsel^T[mm][x]
  const float* irow = bti + (size_t)col * NN;
#pragma unroll 4
  for (int k0 = 0; k0 < NN; k0 += 4) {
    int ka = k0 + 2 * half;
    v2f a   = *(const v2f*)(arow + ka);
    v2f bre = *(const v2f*)(brow + ka);
    v2f bim = *(const v2f*)(irow + ka);
    accr = WMMA_F32(a, bre, accr);
    acci = WMMA_F32(a, bim, acci);
  }
  // transposed store: F^T[img][mm][r]; rows are consecutive over the VGPR index
  int img = m0 >> 7, r0 = (m0 & 127) + 8 * half;
  float* pr = ftr + (size_t)img * NMP * NN + (size_t)col * NN + r0;
  float* pi = fti + (size_t)img * NMP * NN + (size_t)col * NN + r0;
  *(float4*)(pr)     = make_float4(accr[0], accr[1], accr[2], accr[3]);
  *(float4*)(pr + 4) = make_float4(accr[4], accr[5], accr[6], accr[7]);
  *(float4*)(pi)     = make_float4(acci[0], acci[1], acci[2], acci[3]);
  *(float4*)(pi + 4) = make_float4(acci[4], acci[5], acci[6], acci[7]);
}

// ----------------------------------- stage 2: Inm^T = (A_parity @ F)^T (complex)
__global__ void k_stage2(const float* __restrict__ ftr, const float* __restrict__ fti,
                         const float* __restrict__ aer, const float* __restrict__ aei,
                         const float* __restrict__ aor_, const float* __restrict__ aoi,
                         float* __restrict__ itr, float* __restrict__ iti) {
  // grid = dim3(8, 512): blockIdx.x = M-tile over n_idx (128 padded), blockIdx.y = img.
  // 8 waves: wave w -> parity p = w>>2, N-tile nt = w&3 (j = column/2).
  int wave = threadIdx.x >> 5, lane = threadIdx.x & 31;
  int half = lane >> 4, lm = lane & 15;
  int p = wave >> 2, nt = wave & 3;
  int img = blockIdx.y;
  int row0 = blockIdx.x * 16;                  // n_idx tile
  int j = nt * 16 + lm;                        // 0..63
  int mm = 2 * j + p;                          // 0..127 (127 = pad col, zeros)
  const float* ar = ((p == 0) ? aer : aor_) + (size_t)(row0 + lm) * NN;
  const float* ai = ((p == 0) ? aei : aoi) + (size_t)(row0 + lm) * NN;
  const float* fr = ftr + (size_t)img * NMP * NN + (size_t)mm * NN; // F^T[img][mm][r]
  const float* fi = fti + (size_t)img * NMP * NN + (size_t)mm * NN;
  v8f accr = {}, acci = {};
#pragma unroll 4
  for (int k0 = 0; k0 < NN; k0 += 4) {
    int ka = k0 + 2 * half;
    v2f are = *(const v2f*)(ar + ka);
    v2f aim = *(const v2f*)(ai + ka);
    v2f bre = *(const v2f*)(fr + ka);
    v2f bim = *(const v2f*)(fi + ka);
    v2f nbim = -bim;            // f32 WMMA NEG only allows CNeg -> negate in VALU
    accr = WMMA_F32(are, bre, accr);
    accr = WMMA_F32(aim, nbim, accr);
    acci = WMMA_F32(are, bim, acci);
    acci = WMMA_F32(aim, bre, acci);
  }
  // transposed store: Inm^T[img][mm][n_idx]
  int r0 = row0 + 8 * half;
  float* pr = itr + (size_t)img * NMP * NMP + (size_t)mm * NMP + r0;
  float* pi = iti + (size_t)img * NMP * NMP + (size_t)mm * NMP + r0;
  *(float4*)(pr)     = make_float4(accr[0], accr[1], accr[2], accr[3]);
  *(float4*)(pr + 4) = make_float4(accr[4], accr[5], accr[6], accr[7]);
  *(float4*)(pi)     = make_float4(acci[0], acci[1], acci[2], acci[3]);
  *(float4*)(pi + 4) = make_float4(acci[4], acci[5], acci[6], acci[7]);
}

// ------------------------------------------------ fold: J[mm][img][n] (gathered)
__global__ void k_fold(const float* __restrict__ itr, const float* __restrict__ iti,
                       float* __restrict__ jre, float* __restrict__ jim) {
  size_t idx = (size_t)blockIdx.x * blockDim.x + threadIdx.x;
  if (idx >= SZ_J) return;
  int n   = (int)(idx & 63);
  int img = (int)((idx >> 6) & 511);
  int mm  = (int)(idx >> 15);                  // < 127
  const float* Ire = itr + (size_t)img * NMP * NMP + (size_t)mm * NMP;
  const float* Iim = iti + (size_t)img * NMP * NMP + (size_t)mm * NMP;
  float s = (mm & 1) ? -1.f : 1.f;             // (-1)^mm  (spin=1, m = mm-63)
  float re = Ire[n];
  float im = Iim[n];
  if (n >= 1) {
    re += s * Ire[127 - n];
    im += s * Iim[127 - n];
  }
  jre[idx] = re;
  jim[idx] = im;
}

// ------------------------- stage 3: S = J_mm @ D_mm^T over image batch, apply fc
__global__ void k_stage3(const float* __restrict__ jre, const float* __restrict__ jim,
                         const float* __restrict__ dt,
                         const float* __restrict__ fcre, const float* __restrict__ fcim,
                         float* __restrict__ out) {
  // grid = dim3(32, 127): blockIdx.x = img-tile, blockIdx.y = mm. 4 waves = l-tiles.
  int wave = threadIdx.x >> 5, lane = threadIdx.x & 31;
  int half = lane >> 4, lm = lane & 15;
  int mm = blockIdx.y;
  int i0 = blockIdx.x * 16;
  int l0 = wave * 16;
  const float* jr = jre + ((size_t)mm * NIMG + (i0 + lm)) * NL;
  const float* ji = jim + ((size_t)mm * NIMG + (i0 + lm)) * NL;
  const float* dcol = dt + ((size_t)mm * NL + (l0 + lm)) * NL;  // D^T[mm][l][n]
  v8f accr = {}, acci = {};
#pragma unroll 4
  for (int k0 = 0; k0 < NL; k0 += 4) {
    int ka = k0 + 2 * half;
    v2f are = *(const v2f*)(jr + ka);
    v2f aim = *(const v2f*)(ji + ka);
    v2f b   = *(const v2f*)(dcol + ka);
    accr = WMMA_F32(are, b, accr);
    acci = WMMA_F32(aim, b, acci);
  }
  int l = l0 + lm;
  float frc = fcre[(size_t)l * NM + mm];
  float fic = fcim[(size_t)l * NM + mm];
#pragma unroll
  for (int v = 0; v < 8; ++v) {
    int img = i0 + v + 8 * half;
    float sr = accr[v], si = acci[v];
    size_t o = (((size_t)img * NL + l) * NM + mm) * 2;
    out[o]     = frc * sr - fic * si;
    out[o + 1] = frc * si + fic * sr;
  }
}

// ---------------------------------------------------------------------------
extern "C" void kernel_launch(void* const* d_in, const int* in_sizes, int n_in,
                              void* d_out, int out_size, void* d_ws, size_t ws_size,
                              hipStream_t stream) {
  (void)in_sizes; (void)n_in; (void)out_size; (void)ws_size;
  const float* sph  = (const float*)d_in[0];
  const float* qw   = (const float*)d_in[1];
  const float* wd   = (const float*)d_in[2];
  const float* fcre = (const float*)d_in[3];
  const float* fcim = (const float*)d_in[4];
  float* out = (float*)d_out;

  float* ws = (float*)d_ws;        // needs ~170 MB of workspace
  size_t off = 0;
  float* Btr = ws + off;  off += SZ_B;
  float* Bti = ws + off;  off += SZ_B;
  float* Aer = ws + off;  off += SZ_A;
  float* Aei = ws + off;  off += SZ_A;
  float* Aor = ws + off;  off += SZ_A;
  float* Aoi = ws + off;  off += SZ_A;
  float* Dt  = ws + off;  off += SZ_DT;
  float* Ftr = ws + off;  off += SZ_F;
  float* Fti = ws + off;  off += SZ_F;
  float* Itr = ws + off;  off += SZ_I;
  float* Iti = ws + off;  off += SZ_I;
  float* Jre = ws + off;  off += SZ_J;
  float* Jim = ws + off;  off += SZ_J;

  k_build_b<<<(int)((SZ_B + 255) / 256), 256, 0, stream>>>(Btr, Bti);
  k_build_a<<<(int)((SZ_A + 255) / 256), 256, 0, stream>>>(qw, Aer, Aei, Aor, Aoi);
  k_build_dt<<<(int)((SZ_DT + 255) / 256), 256, 0, stream>>>(wd, Dt);

  k_stage1<<<4096, 256, 0, stream>>>(sph, Btr, Bti, Ftr, Fti);
  k_stage2<<<dim3(8, 512), 256, 0, stream>>>(Ftr, Fti, Aer, Aei, Aor, Aoi, Itr, Iti);
  k_fold<<<(int)((SZ_J + 255) / 256), 256, 0, stream>>>(Itr, Iti, Jre, Jim);
  k_stage3<<<dim3(32, 127), 128, 0, stream>>>(Jre, Jim, Dt, fcre, fcim, out);
}